// FuseRec_SPEX_9096740733362
// MI455X (gfx1250) — compile-verified
//
#include <hip/hip_runtime.h>
#include <hip/hip_bf16.h>
#include <math.h>
#include <stdint.h>

// ---------------- problem constants ----------------
#define BB   4096
#define LL   50
#define KK   10
#define LII  20
#define DD   64
#define CC   10
#define NID  50001   // NI + 1
#define NUD  100001  // NU + 1
#define G4   256     // 4*D gate width
#define KP   80      // padded K stride for item-table tile (74 -> 80)

typedef __attribute__((ext_vector_type(2))) float v2f;
typedef __attribute__((ext_vector_type(8))) float v8f;

__device__ __forceinline__ v8f wmma4(v2f a, v2f b, v8f c) {
  // D = A(16x4) * B(4x16) + C(16x16), full f32
  return __builtin_amdgcn_wmma_f32_16x16x4_f32(false, a, false, b, (short)0, c,
                                               false, false);
}

// CDNA5 async global->LDS copy (16B per lane), tracked by ASYNCcnt.
__device__ __forceinline__ void async_copy_b128(uint32_t lds_off, const void* src) {
  asm volatile("global_load_async_to_lds_b128 %0, %1, off"
               :: "v"(lds_off), "v"(src) : "memory");
}
__device__ __forceinline__ void wait_async0() {
  asm volatile("s_wait_asynccnt 0" ::: "memory");
}
__device__ __forceinline__ uint32_t lds_off32(const void* p) {
  // generic shared address: low 32 bits == LDS byte offset (aperture truncation)
  return (uint32_t)(uintptr_t)p;
}

__device__ __forceinline__ float sigf(float x) { return 1.0f / (1.0f + expf(-x)); }
__device__ __forceinline__ float lrelu(float x) { return x > 0.0f ? x : 0.01f * x; }

__device__ __forceinline__ float blockReduce64(float v, float* red, int d) {
  red[d] = v;
  __syncthreads();
#pragma unroll
  for (int s = 32; s > 0; s >>= 1) {
    if (d < s) red[d] += red[d + s];
    __syncthreads();
  }
  float r = red[0];
  __syncthreads();
  return r;
}

// ---------------- kernel: zero scratch ----------------
__global__ void k_zero(float* p, int n) {
  int i = blockIdx.x * blockDim.x + threadIdx.x;
  if (i < n) p[i] = 0.0f;
}

// ---------------- kernel 1: ju_all / jv_all = concat(item_emb,i_class)@W + b
// grid: (ceil(NID/16), 2), block 128 (4 waves, wave = N-tile of 16; N=64)
__global__ void __launch_bounds__(128)
k_jall(const float* __restrict__ item_emb,
       const float* __restrict__ i_class,
       const float* __restrict__ W1, const float* __restrict__ b1,
       const float* __restrict__ W3, const float* __restrict__ b3,
       float* __restrict__ ju_all, float* __restrict__ jv_all) {
  __shared__ float sA[16 * KP];
  const int tile = blockIdx.x;
  const float* W    = blockIdx.y ? W3 : W1;
  const float* bias = blockIdx.y ? b3 : b1;
  float* dst        = blockIdx.y ? jv_all : ju_all;
  const int tid = threadIdx.x;
  // stage A tile: 16 rows x 80 cols (0-63 item_emb, 64-73 i_class, 74-79 zero)
  for (int e = tid; e < 16 * KP; e += 128) {
    int r = e / KP, cc = e % KP;
    int row = tile * 16 + r;
    float v = 0.0f;
    if (row < NID) {
      if (cc < DD) v = item_emb[(size_t)row * DD + cc];
      else if (cc < DD + CC) v = i_class[(size_t)row * CC + (cc - DD)];
    }
    sA[e] = v;
  }
  __syncthreads();
  const int wave = tid >> 5;
  const int lane = tid & 31;
  const int m = lane & 15, hi = lane >> 4;
  const int ncol = wave * 16 + m;
  v8f acc = {};
  for (int k0 = 0; k0 < 76; k0 += 4) {   // K = 74 padded to 76
    const int k = k0 + 2 * hi;           // even pair start
    v2f a = *(const v2f*)&sA[m * KP + k];
    v2f b;
#pragma unroll
    for (int j = 0; j < 2; ++j) {
      int k1 = k + j;
      int kc = (k1 < 74) ? k1 : 0;       // safe clamped index
      float bv = W[kc * DD + ncol];
      b[j] = (k1 < 74) ? bv : 0.0f;      // branchless zero for K tail
    }
    acc = wmma4(a, b, acc);
  }
#pragma unroll
  for (int r = 0; r < 8; ++r) {
    int rowD = tile * 16 + r + 8 * hi;
    if (rowD < NID) dst[(size_t)rowD * DD + ncol] = acc[r] + bias[ncol];
  }
}

// ---------------- kernel 2: one LSTM step (fused [x|h] @ [Wih;Whh], K=128)
// grid: B/16 blocks, block 256 (8 waves, each does 2 N-tiles of the 256 gates)
__global__ void __launch_bounds__(256)
k_lstm_step(const float* __restrict__ ju_all,
            const int* __restrict__ u_items,
            const int* __restrict__ u_items_mask,
            const float* __restrict__ Wih,
            const float* __restrict__ Whh,
            const float* __restrict__ bih,
            const float* __restrict__ bhh,
            float* __restrict__ h, float* __restrict__ c,
            float* __restrict__ hu, int t) {
  __shared__ float sA[16 * 128];
  __shared__ float sG[16 * G4];
  const int tile = blockIdx.x;
  const int tid = threadIdx.x;
  // stage A = [x | h] for 16 batch rows via async global->LDS b128 copies
#pragma unroll
  for (int e4 = tid; e4 < 16 * 32; e4 += 256) {     // 32 float4 per row
    int r = e4 >> 5, c4 = (e4 & 31) * 4;
    int row = tile * 16 + r;
    int idx = u_items[row * LL + t];
    const float* src = (c4 < DD)
        ? &ju_all[(size_t)idx * DD + c4]
        : &h[(size_t)row * DD + (c4 - DD)];
    async_copy_b128(lds_off32(&sA[r * 128 + c4]), src);
  }
  wait_async0();
  __syncthreads();
  const int wave = tid >> 5, lane = tid & 31;
  const int m = lane & 15, hi = lane >> 4;
#pragma unroll
  for (int nt = 0; nt < 2; ++nt) {
    const int ntile = wave * 2 + nt;
    const int ncol = ntile * 16 + m;
    v8f acc = {};
    for (int k0 = 0; k0 < 128; k0 += 4) {
      const int k = k0 + 2 * hi;                     // even pair start
      v2f a = *(const v2f*)&sA[m * 128 + k];         // ds_load_b64
      // branchless base select: pairs never straddle k=64
      const float* pw = (k < DD) ? (Wih + (size_t)k * G4)
                                 : (Whh + (size_t)(k - DD) * G4);
      v2f b;
      b[0] = pw[ncol];
      b[1] = pw[G4 + ncol];
      acc = wmma4(a, b, acc);
    }
#pragma unroll
    for (int r = 0; r < 8; ++r) sG[(r + 8 * hi) * G4 + ncol] = acc[r];
  }
  __syncthreads();
  // elementwise gate update
  for (int e = tid; e < 16 * DD; e += 256) {
    int r = e >> 6, d = e & 63;
    int row = tile * 16 + r;
    float gi = sG[r * G4 + d]          + bih[d]          + bhh[d];
    float gf = sG[r * G4 + DD + d]     + bih[DD + d]     + bhh[DD + d];
    float gg = sG[r * G4 + 2 * DD + d] + bih[2 * DD + d] + bhh[2 * DD + d];
    float go = sG[r * G4 + 3 * DD + d] + bih[3 * DD + d] + bhh[3 * DD + d];
    float cold = c[(size_t)row * DD + d];
    float cn = sigf(gf) * cold + sigf(gi) * tanhf(gg);
    float hn = sigf(go) * tanhf(cn);
    c[(size_t)row * DD + d] = cn;
    h[(size_t)row * DD + d] = hn;
    if (u_items_mask[row] - 1 == t) hu[(size_t)row * DD + d] = hn;
  }
}

// ---------------- kernel 3: out = concat(z, i_e, z*i_e) @ W + b   (K=192,N=64)
// grid: B/16 blocks, block 128 (4 waves, wave = N-tile)
__global__ void __launch_bounds__(128)
k_fuse3(const float* __restrict__ z, const int* __restrict__ items,
        const float* __restrict__ item_emb,
        const float* __restrict__ W, const float* __restrict__ bias,
        float* __restrict__ out) {
  __shared__ float sA[16 * 192];
  const int tile = blockIdx.x;
  const int tid = threadIdx.x;   // 128
  // stage cols 0-63 (z) and 64-127 (i_e) via async global->LDS b128 copies
#pragma unroll
  for (int e4 = tid; e4 < 16 * 32; e4 += 128) {     // 32 float4 over 128 cols
    int r = e4 >> 5, c4 = (e4 & 31) * 4;
    int row = tile * 16 + r;
    int it = items[row];
    const float* src = (c4 < DD)
        ? &z[(size_t)row * DD + c4]
        : &item_emb[(size_t)it * DD + (c4 - DD)];
    async_copy_b128(lds_off32(&sA[r * 192 + c4]), src);
  }
  wait_async0();
  __syncthreads();
  // cols 128-191: product z * i_e from staged LDS
  for (int e = tid; e < 16 * DD; e += 128) {
    int r = e >> 6, d = e & 63;
    sA[r * 192 + 2 * DD + d] = sA[r * 192 + d] * sA[r * 192 + DD + d];
  }
  __syncthreads();
  const int wave = tid >> 5, lane = tid & 31;
  const int m = lane & 15, hi = lane >> 4;
  const int ncol = wave * 16 + m;
  v8f acc = {};
  for (int k0 = 0; k0 < 192; k0 += 4) {
    const int k = k0 + 2 * hi;
    v2f a = *(const v2f*)&sA[m * 192 + k];           // ds_load_b64
    v2f b;
    b[0] = W[(size_t)k * DD + ncol];
    b[1] = W[(size_t)(k + 1) * DD + ncol];
    acc = wmma4(a, b, acc);
  }
#pragma unroll
  for (int r = 0; r < 8; ++r) {
    int row = tile * 16 + r + 8 * hi;
    out[(size_t)row * DD + ncol] = acc[r] + bias[ncol];
  }
}

// ---------------- kernel 4: social attention -> su   (B blocks x 64 threads)
__global__ void __launch_bounds__(64)
k_social(const int* __restrict__ users, const int* __restrict__ u_frids,
         const int* __restrict__ u_frids_mask,
         const int* __restrict__ u_frids_items,
         const float* __restrict__ user_emb,
         const float* __restrict__ jv_all,
         const float* __restrict__ W4, const float* __restrict__ b4,
         float* __restrict__ su) {
  const int b = blockIdx.x, d = threadIdx.x;
  __shared__ float red[64];
  __shared__ float sat[KK], sauv[KK];
  float u_d = user_emb[(size_t)users[b] * DD + d];
  float uW = blockReduce64(u_d * W4[d], red, d);
  int fmask = u_frids_mask[b];
  for (int kf = 0; kf < KK; ++kf) {
    float vd = user_emb[(size_t)u_frids[b * KK + kf] * DD + d];
    float vW = blockReduce64(vd * W4[DD + d], red, d);
    if (d == 0) sat[kf] = lrelu(uW + vW + b4[0]);
  }
  __syncthreads();
  if (d == 0) {
    float mx = -INFINITY;
    for (int kf = 0; kf < fmask; ++kf) mx = fmaxf(mx, sat[kf]);
    float sm = 0.0f;
    for (int kf = 0; kf < fmask; ++kf) { float e = expf(sat[kf] - mx); sauv[kf] = e; sm += e; }
    for (int kf = 0; kf < KK; ++kf) sauv[kf] = (kf < fmask) ? sauv[kf] / sm : 0.0f;
  }
  __syncthreads();
  float acc = 0.0f;
  float inv = 1.0f / (float)fmask;
  for (int kf = 0; kf < KK; ++kf) {
    float w = sauv[kf];
    if (w != 0.0f) {
      float pv = 0.0f;
      for (int li = 0; li < LII; ++li) {
        int idx = u_frids_items[(b * KK + kf) * LII + li];
        pv += jv_all[(size_t)idx * DD + d];
      }
      acc += pv * inv * w;
    }
  }
  su[(size_t)b * DD + d] = acc;
}

// ---------------- kernel 5: s1/s2 scores      (B blocks x 64 threads)
__global__ void __launch_bounds__(64)
k_s12(const int* __restrict__ items, const int* __restrict__ F_i,
      const float* __restrict__ item_emb,
      const float* __restrict__ W6, const float* __restrict__ b6,
      float* __restrict__ s1, float* __restrict__ s2) {
  const int b = blockIdx.x, d = threadIdx.x;
  __shared__ float red[64];
  float ie = item_emb[(size_t)items[b] * DD + d];
  float iW = blockReduce64(ie * W6[d], red, d);
  for (int s = 0; s < 2; ++s) {
    float* dst = s ? s2 : s1;
    for (int kf = 0; kf < KK; ++kf) {
      int idx = F_i[b * 2 * KK + s * KK + kf];
      float fv = item_emb[(size_t)idx * DD + d];
      float fW = blockReduce64(fv * W6[DD + d], red, d);
      if (d == 0) dst[b * KK + kf] = lrelu(iW + fW + b6[0]);
    }
  }
}

// ---------------- kernel 6: per-column (batch-axis) softmax stats, 20 blocks
__global__ void __launch_bounds__(256)
k_colstats(const float* __restrict__ s1, const float* __restrict__ s2,
           float* __restrict__ red) {
  const int col = blockIdx.x;            // 0..9 -> s1, 10..19 -> s2
  const float* s = (col < KK) ? s1 : s2;
  const int kf = col % KK;
  __shared__ float sh[256];
  const int tid = threadIdx.x;
  float mx = -INFINITY;
  for (int row = tid; row < BB; row += 256) mx = fmaxf(mx, s[row * KK + kf]);
  sh[tid] = mx;
  __syncthreads();
  for (int st = 128; st > 0; st >>= 1) {
    if (tid < st) sh[tid] = fmaxf(sh[tid], sh[tid + st]);
    __syncthreads();
  }
  float m = sh[0];
  __syncthreads();
  float sm = 0.0f;
  for (int row = tid; row < BB; row += 256) sm += expf(s[row * KK + kf] - m);
  sh[tid] = sm;
  __syncthreads();
  for (int st = 128; st > 0; st >>= 1) {
    if (tid < st) sh[tid] += sh[tid + st];
    __syncthreads();
  }
  if (tid == 0) { red[col * 2] = m; red[col * 2 + 1] = sh[0]; }
}

// ---------------- kernel 7: yi blend            (B blocks x 64 threads)
__global__ void __launch_bounds__(64)
k_yi(const int* __restrict__ F_i, const float* __restrict__ item_emb,
     const float* __restrict__ s1, const float* __restrict__ s2,
     const float* __restrict__ red, const float* __restrict__ alpha,
     float* __restrict__ yi) {
  const int b = blockIdx.x, d = threadIdx.x;
  float a = alpha[0];
  float acc1 = 0.0f, acc2 = 0.0f;
  for (int kf = 0; kf < KK; ++kf) {
    float w1 = expf(s1[b * KK + kf] - red[kf * 2]) / red[kf * 2 + 1];
    float w2 = expf(s2[b * KK + kf] - red[(KK + kf) * 2]) / red[(KK + kf) * 2 + 1];
    acc1 += w1 * item_emb[(size_t)F_i[b * 2 * KK + kf] * DD + d];
    acc2 += w2 * item_emb[(size_t)F_i[b * 2 * KK + KK + kf] * DD + d];
  }
  yi[(size_t)b * DD + d] = a * acc1 + (1.0f - a) * acc2;
}

// ---------------- kernel 8: final dots + sigmoid (wave per row)
__global__ void __launch_bounds__(256)
k_final(const float* __restrict__ hu, const float* __restrict__ hui,
        const float* __restrict__ su, const float* __restrict__ sui,
        const float* __restrict__ yi, const float* __restrict__ lambdas,
        float* __restrict__ out) {
  const int wid = threadIdx.x >> 5, lane = threadIdx.x & 31;
  const int row = blockIdx.x * 8 + wid;
  const float l0 = lambdas[0], l1 = lambdas[1], l2 = lambdas[2], l3 = lambdas[3];
  float p = 0.0f;
#pragma unroll
  for (int j = 0; j < 2; ++j) {
    int e = row * DD + lane + 32 * j;
    float y = yi[e];
    p += (l0 * hu[e] + l1 * hui[e] + l2 * su[e] + l3 * sui[e]) * y;
  }
#pragma unroll
  for (int m = 16; m > 0; m >>= 1) p += __shfl_xor(p, m, 32);
  if (lane == 0) out[row] = sigf(p);
}

// ---------------- launch ----------------
extern "C" void kernel_launch(void* const* d_in, const int* in_sizes, int n_in,
                              void* d_out, int out_size, void* d_ws, size_t ws_size,
                              hipStream_t stream) {
  const int*   users        = (const int*)d_in[0];
  const int*   items        = (const int*)d_in[1];
  const int*   u_items      = (const int*)d_in[2];
  const int*   u_items_mask = (const int*)d_in[3];
  const int*   u_frids      = (const int*)d_in[4];
  const int*   u_frids_mask = (const int*)d_in[5];
  const int*   u_frids_items= (const int*)d_in[6];
  const int*   F_i          = (const int*)d_in[7];
  const float* user_emb     = (const float*)d_in[8];
  const float* item_emb     = (const float*)d_in[9];
  const float* i_class      = (const float*)d_in[10];
  const float* W1  = (const float*)d_in[11];
  const float* b1  = (const float*)d_in[12];
  const float* Wih = (const float*)d_in[13];
  const float* Whh = (const float*)d_in[14];
  const float* bih = (const float*)d_in[15];
  const float* bhh = (const float*)d_in[16];
  const float* W2  = (const float*)d_in[17];
  const float* b2  = (const float*)d_in[18];
  const float* W3  = (const float*)d_in[19];
  const float* b3  = (const float*)d_in[20];
  const float* W4  = (const float*)d_in[21];
  const float* b4  = (const float*)d_in[22];
  const float* W5  = (const float*)d_in[23];
  const float* b5  = (const float*)d_in[24];
  const float* W6  = (const float*)d_in[25];
  const float* b6  = (const float*)d_in[26];
  const float* alpha   = (const float*)d_in[27];
  const float* lambdas = (const float*)d_in[28];
  float* out = (float*)d_out;

  // workspace layout (floats)
  float* ws = (float*)d_ws;
  const size_t sz_jall = (size_t)NID * DD;   // 3,200,064
  const size_t sz_bd   = (size_t)BB * DD;    //   262,144
  float* ju_all = ws;                        size_t off = sz_jall;
  float* jv_all = ws + off;                  off += sz_jall;
  float* h      = ws + off;                  off += sz_bd;
  float* c      = ws + off;                  off += sz_bd;
  float* hu     = ws + off;                  off += sz_bd;
  float* hui    = ws + off;                  off += sz_bd;
  float* su     = ws + off;                  off += sz_bd;
  float* sui    = ws + off;                  off += sz_bd;
  float* yi     = ws + off;                  off += sz_bd;
  float* s1     = ws + off;                  off += (size_t)BB * KK;
  float* s2     = ws + off;                  off += (size_t)BB * KK;
  float* red    = ws + off;                  off += 40;

  // 0) zero h, c (adjacent in ws)
  {
    int n = 2 * (int)sz_bd;
    k_zero<<<(n + 255) / 256, 256, 0, stream>>>(h, n);
  }
  // 1) item-table GEMMs: ju_all (W1), jv_all (W3)
  k_jall<<<dim3((NID + 15) / 16, 2), 128, 0, stream>>>(item_emb, i_class, W1, b1,
                                                       W3, b3, ju_all, jv_all);
  // 2) LSTM recurrence, 50 steps
  for (int t = 0; t < LL; ++t) {
    k_lstm_step<<<BB / 16, 256, 0, stream>>>(ju_all, u_items, u_items_mask, Wih,
                                             Whh, bih, bhh, h, c, hu, t);
  }
  // 3) hui = concat(hu, i_e, hu*i_e) @ W2 + b2
  k_fuse3<<<BB / 16, 128, 0, stream>>>(hu, items, item_emb, W2, b2, hui);
  // 4) social attention -> su
  k_social<<<BB, 64, 0, stream>>>(users, u_frids, u_frids_mask, u_frids_items,
                                  user_emb, jv_all, W4, b4, su);
  // 5) sui = concat(su, i_e, su*i_e) @ W5 + b5
  k_fuse3<<<BB / 16, 128, 0, stream>>>(su, items, item_emb, W5, b5, sui);
  // 6) s1/s2 scores
  k_s12<<<BB, 64, 0, stream>>>(items, F_i, item_emb, W6, b6, s1, s2);
  // 7) batch-axis softmax stats (20 columns)
  k_colstats<<<20, 256, 0, stream>>>(s1, s2, red);
  // 8) yi blend
  k_yi<<<BB, 64, 0, stream>>>(F_i, item_emb, s1, s2, red, alpha, yi);
  // 9) final scores
  k_final<<<BB / 8, 256, 0, stream>>>(hu, hui, su, sui, yi, lambdas, out);
}